// PerImageLABNormalizer_50551765074339
// MI455X (gfx1250) — compile-verified
//
#include <hip/hip_runtime.h>
#include <stdint.h>

typedef float v4f __attribute__((ext_vector_type(4)));

#define BATCH  64
#define CHN    3
#define HW     (512 * 512)          // 262144 elements per (batch, channel)
#define NBINS  100
#define HPAD   128                  // padded histogram row
#define NJOBS  (BATCH * CHN)        // 192
#define SPLIT  4                    // blocks per job in histogram pass
#define PART   (HW / SPLIT)         // 65536 elements per block
#define TPB    256
#define TILES  (PART / (TPB * 4))   // 64 float4-tiles per block

// ---------------------------------------------------------------------------
// Pass 1: per-(batch,channel) histograms.
// Streams pixels through LDS with double-buffered async DMA
// (global_load_async_to_lds_b128), per-wave private LDS histograms,
// one global atomicAdd per bin per block at the end.
// Binning matches the reference bit-for-bit: fp32 (x - lo) / bin_size,
// truncation (== floor for non-negative), clip to NBINS-1.
// ---------------------------------------------------------------------------
__global__ __launch_bounds__(TPB) void hist_kernel(const float* __restrict__ lab,
                                                   unsigned* __restrict__ hist) {
    __shared__ unsigned h[8][HPAD];     // 8 waves * 128 bins = 4 KB
    __shared__ v4f tile[2][TPB];        // ping-pong DMA tiles, 8 KB

    const int tid  = threadIdx.x;
    const int wave = tid >> 5;
    const int job  = blockIdx.x / SPLIT;
    const int part = blockIdx.x % SPLIT;
    const int c    = job % CHN;

    for (int i = tid; i < 8 * HPAD; i += TPB) ((unsigned*)h)[i] = 0u;
    __syncthreads();

    const float lo      = (c == 0) ? 0.0f   : -128.0f;
    const float hi      = (c == 0) ? 100.0f :  127.0f;
    const float binSize = (hi - lo) / (float)NBINS;   // fp32, same as reference

    const float*   base  = lab + (size_t)job * HW + (size_t)part * PART;
    const uint64_t saddr = (uint64_t)(uintptr_t)base;

    // Generic pointer to __shared__ carries the LDS byte offset in its low
    // 32 bits (flat shared-aperture encoding) -> usable as DMA VDST offset.
    unsigned lds0 = (unsigned)(uintptr_t)&tile[0][tid];
    unsigned lds1 = (unsigned)(uintptr_t)&tile[1][tid];

    // Prime the pipeline: tile 0 in flight.
    {
        unsigned voff = (unsigned)(tid * 16);
        asm volatile("global_load_async_to_lds_b128 %0, %1, %2"
                     :: "v"(lds0), "v"(voff), "s"(saddr) : "memory");
    }

    for (int t = 0; t < TILES; ++t) {
        if (t + 1 < TILES) {
            unsigned loff = ((t + 1) & 1) ? lds1 : lds0;
            unsigned voff = (unsigned)((((t + 1) * TPB) + tid) * 16);
            asm volatile("global_load_async_to_lds_b128 %0, %1, %2"
                         :: "v"(loff), "v"(voff), "s"(saddr) : "memory");
            asm volatile("s_wait_asynccnt 1" ::: "memory");   // tile t landed
        } else {
            asm volatile("s_wait_asynccnt 0" ::: "memory");   // last tile landed
        }

        v4f x = tile[t & 1][tid];
#pragma unroll
        for (int k = 0; k < 4; ++k) {
            float v = x[k];
            if (v >= lo && v <= hi) {
                int b = (int)((v - lo) / binSize);  // >= 0, trunc == floor
                b = b > (NBINS - 1) ? (NBINS - 1) : b;
                atomicAdd(&h[wave][b], 1u);         // ds_add_u32, wave-private
            }
        }
    }
    __syncthreads();

    if (tid < NBINS) {
        unsigned s = 0;
#pragma unroll
        for (int w = 0; w < 8; ++w) s += h[w][tid];
        atomicAdd(&hist[job * HPAD + tid], s);
    }
}

// ---------------------------------------------------------------------------
// Pass 2: tiny. One thread per (batch, channel): scan 100 bins, reproduce
// jax top_k / argmax tie semantics (first occurrence wins), emit peak value.
// ---------------------------------------------------------------------------
__global__ void peak_kernel(const unsigned* __restrict__ hist,
                            float* __restrict__ peaks) {
    int job = blockIdx.x * blockDim.x + threadIdx.x;
    if (job >= NJOBS) return;
    const int c = job % CHN;
    const unsigned* h = hist + job * HPAD;

    int m1 = (int)h[0], i1 = 0;
    int m2 = -1,        i2 = 0;
    for (int i = 1; i < NBINS; ++i) {
        int v = (int)h[i];
        if (v > m1)      { m2 = m1; i2 = i1; m1 = v; i1 = i; }
        else if (v > m2) { m2 = v;  i2 = i; }
    }
    const float lo = (c == 0) ? 0.0f   : -128.0f;
    const float hi = (c == 0) ? 100.0f :  127.0f;
    const float bs = (hi - lo) / (float)NBINS;
    const int pick = (c == 0) ? i2 : i1;       // L: second peak; A/B: primary
    peaks[job] = lo + ((float)pick + 0.5f) * bs;
}

// ---------------------------------------------------------------------------
// Pass 3: streaming normalize, one float4 per thread, non-temporal b128.
// Exact reference arithmetic: ((x * ref) / peak + K) / D, all fp32.
// ---------------------------------------------------------------------------
__global__ __launch_bounds__(256) void norm_kernel(const float* __restrict__ lab,
                                                   const float* __restrict__ refl,
                                                   const float* __restrict__ refa,
                                                   const float* __restrict__ refb,
                                                   const float* __restrict__ peaks,
                                                   float* __restrict__ out) {
    const int f = blockIdx.x * blockDim.x + threadIdx.x;   // float4 index
    const int total = NJOBS * (HW / 4);
    if (f >= total) return;

    const int job = f >> 16;          // HW/4 == 65536 float4 per channel
    const int c   = job % CHN;

    const float ref = (c == 0) ? refl[0] : (c == 1) ? refa[0] : refb[0];
    const float p   = peaks[job];
    const float add = (c == 0) ? 100.0f : 128.0f;
    const float div = (c == 0) ? 200.0f : 255.0f;

    v4f x = __builtin_nontemporal_load((const v4f*)lab + f);
    v4f t = (x * ref) / p;            // matches (x * ref) / peak rounding
    v4f y = (t + add) / div;
    __builtin_nontemporal_store(y, (v4f*)out + f);
}

extern "C" void kernel_launch(void* const* d_in, const int* in_sizes, int n_in,
                              void* d_out, int out_size, void* d_ws, size_t ws_size,
                              hipStream_t stream) {
    const float* lab  = (const float*)d_in[0];
    const float* refl = (const float*)d_in[1];
    const float* refa = (const float*)d_in[2];
    const float* refb = (const float*)d_in[3];
    float* out = (float*)d_out;

    unsigned* hist  = (unsigned*)d_ws;
    float*    peaks = (float*)((char*)d_ws + (size_t)NJOBS * HPAD * sizeof(unsigned));

    hipMemsetAsync(d_ws, 0, (size_t)NJOBS * HPAD * sizeof(unsigned), stream);

    hist_kernel<<<NJOBS * SPLIT, TPB, 0, stream>>>(lab, hist);
    peak_kernel<<<(NJOBS + 63) / 64, 64, 0, stream>>>(hist, peaks);

    const int total_f4 = NJOBS * (HW / 4);                 // 12,582,912
    norm_kernel<<<(total_f4 + 255) / 256, 256, 0, stream>>>(lab, refl, refa, refb,
                                                            peaks, out);
}